// DeepHitLoss_62843961475663
// MI455X (gfx1250) — compile-verified
//
#include <hip/hip_runtime.h>
#include <hip/hip_bf16.h>

// ---------------------------------------------------------------------------
// DeepHit-style loss, factorized O(N^2) compare/masked-add sweep.
//  S_i = sum over { t_j <  t_i  or (t_j==t_i and j>=i) } of exp(r_j)   (risk-set sum, matches stable desc sort)
//  U_i = sum over { t_j >  t_i }                        of exp(r_j)
//  C_i = count of { t_j >  t_i }
//  lik  = sum_{e_i=1} ( r_i - log S_i )
//  rank = sum_{e_i=1} exp(-r_i) * U_i ;  pairs = sum_{e_i=1} C_i
//  out  = -lik/(nev+1e-8) + 0.2 * (pairs>0 ? rank/max(pairs,1) : rank)
// ---------------------------------------------------------------------------

#define TPB   256
#define CHUNK 512   // j-elements staged in LDS per block

typedef float v2f __attribute__((ext_vector_type(2)));
typedef float v8f __attribute__((ext_vector_type(8)));
typedef int   v2i __attribute__((vector_size(8)));   // matches builtin's V2i param

#if defined(__has_builtin)
#  if __has_builtin(__builtin_amdgcn_global_load_async_to_lds_b64)
#    define HAVE_ASYNC_LDS 1
#  endif
#  if __has_builtin(__builtin_amdgcn_s_wait_asynccnt)
#    define HAVE_WAIT_ASYNC 1
#  endif
#  if __has_builtin(__builtin_amdgcn_wmma_f32_16x16x4_f32)
#    define HAVE_WMMA4 1
#  endif
#endif

struct Accum { float lik; float rank; unsigned int pairs; int nev; };

__global__ __launch_bounds__(TPB) void k_init(float* __restrict__ S, float* __restrict__ U,
                                              float* __restrict__ C, Accum* acc, int n) {
    int i = blockIdx.x * TPB + threadIdx.x;
    if (i < n) { S[i] = 0.0f; U[i] = 0.0f; C[i] = 0.0f; }
    if (i == 0) { acc->lik = 0.0f; acc->rank = 0.0f; acc->pairs = 0u; acc->nev = 0; }
}

__global__ __launch_bounds__(TPB) void k_prep(const float* __restrict__ r, const int* __restrict__ e,
                                              float* __restrict__ er, Accum* acc, int n) {
    int i = blockIdx.x * TPB + threadIdx.x;
    if (i < n) {
        er[i] = expf(r[i]);
        if (e[i] == 1) atomicAdd(&acc->nev, 1);  // int atomics: exact + deterministic
    }
}

// Main O(N^2) sweep: block (x = i-tile of 256, y = j-chunk of 512).
__global__ __launch_bounds__(TPB) void k_pairs(const float* __restrict__ t, const float* __restrict__ er,
                                               float* __restrict__ S, float* __restrict__ U,
                                               float* __restrict__ C, int n) {
    __shared__ __align__(16) float sT[CHUNK];
    __shared__ __align__(16) float sE[CHUNK];
    const int tid    = threadIdx.x;
    const int i      = blockIdx.x * TPB + tid;
    const int jbase  = blockIdx.y * CHUNK;
    const int jcount = (n - jbase < CHUNK) ? (n - jbase) : CHUNK;
    const float ti   = (i < n) ? t[i] : 0.0f;

#if defined(HAVE_ASYNC_LDS)
    if (jcount == CHUNK) {
        const int o = tid * 2;  // 2 floats (8B) per lane per array
        __builtin_amdgcn_global_load_async_to_lds_b64(
            (__attribute__((address_space(1))) v2i*)(t  + jbase + o),
            (__attribute__((address_space(3))) v2i*)(sT + o), 0, 0);
        __builtin_amdgcn_global_load_async_to_lds_b64(
            (__attribute__((address_space(1))) v2i*)(er + jbase + o),
            (__attribute__((address_space(3))) v2i*)(sE + o), 0, 0);
#if defined(HAVE_WAIT_ASYNC)
        __builtin_amdgcn_s_wait_asynccnt(0);
#else
        asm volatile("s_wait_asynccnt 0" ::: "memory");
#endif
    } else
#endif
    {
        for (int k = tid; k < jcount; k += TPB) { sT[k] = t[jbase + k]; sE[k] = er[jbase + k]; }
    }
    __syncthreads();

    float s_le = 0.0f, u_gt = 0.0f, c_gt = 0.0f;
#pragma unroll 4
    for (int k = 0; k < jcount; ++k) {
        const float tj = sT[k];
        const float ej = sE[k];
        const bool gt = (tj > ti);
        const bool le = (tj < ti) || ((tj == ti) && (jbase + k >= i));  // stable-sort tie order
        s_le += le ? ej : 0.0f;
        u_gt += gt ? ej : 0.0f;
        c_gt += gt ? 1.0f : 0.0f;   // <= 8191 per i: exact in fp32
    }
    if (i < n) {
        atomicAdd(&S[i], s_le);
        atomicAdd(&U[i], u_gt);
        atomicAdd(&C[i], c_gt);
    }
}

// Per-i contributions + block reduction. Float partials reduced on the matrix
// pipe: D = ones(16x4) x B(4x16 data) + C accumulated over 4 tiles gives the
// 16 column sums of the 16x16 partial matrix in lanes 0-15 of D's first VGPR
// (layout-invariant total since A is all ones), finished with 4 shfl_xor.
__global__ __launch_bounds__(TPB) void k_reduce(const float* __restrict__ r, const int* __restrict__ e,
                                                const float* __restrict__ S, const float* __restrict__ U,
                                                const float* __restrict__ C, Accum* acc, int n) {
    __shared__ float redA[TPB];
    __shared__ float redB[TPB];
    __shared__ unsigned int redC[TPB];
    const int tid = threadIdx.x;
    const int i   = blockIdx.x * TPB + tid;

    float lik = 0.0f, rank = 0.0f;
    unsigned int cnt = 0u;
    if (i < n && e[i] == 1) {
        lik  = r[i] - logf(S[i]);        // S[i] >= exp(r[i]) > 0 (self term included)
        rank = expf(-r[i]) * U[i];
        cnt  = (unsigned int)(C[i] + 0.5f);
    }
    redA[tid] = lik; redB[tid] = rank; redC[tid] = cnt;
    __syncthreads();

#if defined(HAVE_WMMA4)
    if (tid < 32) {  // whole wave 0 enters -> EXEC all ones for WMMA
        v2f av; av.x = 1.0f; av.y = 1.0f;   // A = ones (16x4)
        v8f cl, cr;
#pragma unroll
        for (int q = 0; q < 8; ++q) { cl[q] = 0.0f; cr[q] = 0.0f; }
#pragma unroll
        for (int g = 0; g < 4; ++g) {       // 4 accumulating WMMAs cover 256 values
            v2f bl; bl.x = redA[g * 64 + tid]; bl.y = redA[g * 64 + 32 + tid];
            cl = __builtin_amdgcn_wmma_f32_16x16x4_f32(false, av, false, bl, (short)0, cl, false, false);
            v2f br; br.x = redB[g * 64 + tid]; br.y = redB[g * 64 + 32 + tid];
            cr = __builtin_amdgcn_wmma_f32_16x16x4_f32(false, av, false, br, (short)0, cr, false, false);
        }
        float tl = cl[0], tr = cr[0];       // lane n (0..15): column sum n
#pragma unroll
        for (int m = 1; m < 16; m <<= 1) {
            tl += __shfl_xor(tl, m, 32);
            tr += __shfl_xor(tr, m, 32);
        }
        unsigned int tc = 0u;
#pragma unroll
        for (int w = 0; w < 8; ++w) tc += redC[tid + 32 * w];
#pragma unroll
        for (int m = 16; m >= 1; m >>= 1) tc += (unsigned int)__shfl_xor((int)tc, m, 32);
        if (tid == 0) {
            atomicAdd(&acc->lik,  tl);
            atomicAdd(&acc->rank, tr);
            atomicAdd(&acc->pairs, tc);
        }
    }
#else
    for (int s = 128; s > 0; s >>= 1) {
        if (tid < s) { redA[tid] += redA[tid + s]; redB[tid] += redB[tid + s]; redC[tid] += redC[tid + s]; }
        __syncthreads();
    }
    if (tid == 0) {
        atomicAdd(&acc->lik, redA[0]);
        atomicAdd(&acc->rank, redB[0]);
        atomicAdd(&acc->pairs, redC[0]);
    }
#endif
}

__global__ void k_out(const Accum* __restrict__ acc, float* __restrict__ out) {
    const float nev      = (float)acc->nev;
    const float lik_loss = -acc->lik / (nev + 1e-8f);
    const float pc       = (float)acc->pairs;
    const float rank_loss = (acc->pairs > 0u) ? (acc->rank / fmaxf(pc, 1.0f)) : acc->rank;
    out[0] = lik_loss + 0.2f * rank_loss;
}

extern "C" void kernel_launch(void* const* d_in, const int* in_sizes, int n_in,
                              void* d_out, int out_size, void* d_ws, size_t ws_size,
                              hipStream_t stream) {
    (void)n_in; (void)out_size; (void)ws_size;
    const float* r = (const float*)d_in[0];
    const float* t = (const float*)d_in[1];
    const int*   e = (const int*)d_in[2];
    const int n = in_sizes[0];

    char*  ws  = (char*)d_ws;
    Accum* acc = (Accum*)ws;
    float* er  = (float*)(ws + 64);
    float* S   = er + n;
    float* U   = S + n;
    float* C   = U + n;

    const int nb      = (n + TPB - 1) / TPB;
    const int jchunks = (n + CHUNK - 1) / CHUNK;

    k_init  <<<nb, TPB, 0, stream>>>(S, U, C, acc, n);
    k_prep  <<<nb, TPB, 0, stream>>>(r, e, er, acc, n);
    dim3 grid(nb, jchunks);
    k_pairs <<<grid, TPB, 0, stream>>>(t, er, S, U, C, n);
    k_reduce<<<nb, TPB, 0, stream>>>(r, e, S, U, C, acc, n);
    k_out   <<<1, 1, 0, stream>>>(acc, (float*)d_out);
}